// Bilinear_41094247088135
// MI455X (gfx1250) — compile-verified
//
#include <hip/hip_runtime.h>

typedef float v8f __attribute__((ext_vector_type(8)));
typedef float v2f __attribute__((ext_vector_type(2)));

// ---------------- compile-time PGA(3,0,1) tables (faithful to reference) ----
namespace pga {
constexpr int BITS_[16] = {0,1,2,4,8,3,5,9,6,10,12,7,11,13,14,15};
constexpr int IDX_[16]  = {0,1,2,5,3,6,8,11,4,7,9,12,10,13,14,15}; // bits -> index
constexpr int pc4(int v){ int c=0; for(int i=0;i<4;i++) c += (v>>i)&1; return c; }
constexpr float scat(int a,int b){ int sw=0; a>>=1; while(a){ sw += pc4(a&b); a>>=1; } return (sw&1)?-1.0f:1.0f; }
struct Tabs {
  float gs[16][16];   // geometric product sign (0 if blade vanishes)
  int   gk[16][16];   // output index (a^b)
  float we[16][16];   // wedge sign (0 unless disjoint)
  float ds[16];       // dual signs
};
constexpr Tabs mk(){
  Tabs t{};
  for(int i=0;i<16;i++){
    t.ds[i] = scat(BITS_[i], 15 ^ BITS_[i]);
    for(int j=0;j<16;j++){
      const int a = BITS_[i], b = BITS_[j], ab = a & b;
      t.gk[i][j] = IDX_[a ^ b];
      t.gs[i][j] = (ab == 0 || !(ab & 1)) ? scat(a,b) : 0.0f;
      t.we[i][j] = (ab == 0) ? scat(a,b) : 0.0f;
    }
  }
  return t;
}
constexpr Tabs T = mk();
} // namespace pga

// ---------------- prep: W[w][C][c] = sum_n w[w][C][c][n]  (4*32*32 floats) --
__global__ void pga_prep_w(const float* __restrict__ w0, const float* __restrict__ w1,
                           const float* __restrict__ w2, const float* __restrict__ w3,
                           float* __restrict__ ws){
  const int e = blockIdx.x * blockDim.x + threadIdx.x;
  if(e >= 4096) return;
  const float* srcs[4] = {w0, w1, w2, w3};
  const float* p = srcs[e >> 10] + (e & 1023) * 9;
  float s = 0.f;
  #pragma unroll
  for(int n = 0; n < 9; n++) s += p[n];
  ws[e] = s;
}

// ---------------- z15 reduction: mean of x[...,15] ---------------------------
__global__ void pga_reduce1(const float* __restrict__ x, float* __restrict__ ws, int n){
  __shared__ float sm[256];
  float s = 0.f;
  for(int i = blockIdx.x * 256 + threadIdx.x; i < n; i += 256 * gridDim.x)
    s += x[(size_t)i * 16 + 15];
  sm[threadIdx.x] = s; __syncthreads();
  #pragma unroll
  for(int off = 128; off > 0; off >>= 1){
    if(threadIdx.x < off) sm[threadIdx.x] += sm[threadIdx.x + off];
    __syncthreads();
  }
  if(threadIdx.x == 0) ws[4097 + blockIdx.x] = sm[0];
}
__global__ void pga_reduce2(float* __restrict__ ws, float n){
  __shared__ float sm[256];
  sm[threadIdx.x] = ws[4097 + threadIdx.x]; __syncthreads();
  #pragma unroll
  for(int off = 128; off > 0; off >>= 1){
    if(threadIdx.x < off) sm[threadIdx.x] += sm[threadIdx.x + off];
    __syncthreads();
  }
  if(threadIdx.x == 0) ws[4096] = sm[0] / n;
}

// ---------------- stage two 32x16 result matrices (D layout -> LDS) ---------
__device__ __forceinline__ void stage_pair(float (*slds)[16][36],
                                           const v8f* accX, const v8f* accY,
                                           int m, int half){
  #pragma unroll
  for(int t = 0; t < 2; t++){
    const int col = 16*t + 8*half;
    float4* px = (float4*)&slds[0][m][col];
    px[0] = make_float4(accX[t][0], accX[t][1], accX[t][2], accX[t][3]);
    px[1] = make_float4(accX[t][4], accX[t][5], accX[t][6], accX[t][7]);
    float4* py = (float4*)&slds[1][m][col];
    py[0] = make_float4(accY[t][0], accY[t][1], accY[t][2], accY[t][3]);
    py[1] = make_float4(accY[t][4], accY[t][5], accY[t][6], accY[t][7]);
  }
}

// ---------------- main kernel ------------------------------------------------
__global__ __launch_bounds__(256) void pga_bilinear_main(
    const float* __restrict__ x, const float* __restrict__ wsbuf,
    float* __restrict__ out, int nTok)
{
  __shared__ float lds[8][2][16][36];          // per-wave transpose slice, pad 36
  const int lane = threadIdx.x & 31;
  const int wave = threadIdx.x >> 5;
  const int half = lane >> 4;
  const int m    = lane & 15;
  float (*slds)[16][36] = lds[wave];

  // Preload W into WMMA A-operand layout: Wa[w][tile][chunk] (b64 loads).
  v2f Wa[4][2][8];
  #pragma unroll
  for(int w = 0; w < 4; w++)
    #pragma unroll
    for(int t = 0; t < 2; t++)
      #pragma unroll
      for(int kk = 0; kk < 8; kk++){
        const float* p = wsbuf + ((w*32 + 16*t + m)*32 + 4*kk + 2*half);
        Wa[w][t][kk] = *(const v2f*)p;
      }
  const float z15 = wsbuf[4096];

  // u = x + x*e0 : per-component partner + sign, via lane bpermute
  constexpr int   SRCL[16] = {0,0,0,0,0,2,3,4,0,0,0,8,9,10,0,14};
  constexpr float CFL [16] = {0.f,1.f,0.f,0.f,0.f,-1.f,-1.f,-1.f,0.f,0.f,0.f,1.f,1.f,1.f,0.f,-1.f};
  const int   bpidx = ((lane & 16) + SRCL[m]) << 2;
  const float cf    = CFL[m];
  const int   base0 = half*32 + m;             // x offset for B-layout load

  const int nWaves = gridDim.x * 8;
  for(int tok = blockIdx.x*8 + wave; tok < nTok; tok += nWaves){
    const float* xt = x + (size_t)tok * 512;
    if(tok + nWaves < nTok)
      __builtin_prefetch(x + (size_t)(tok + nWaves)*512 + lane*16, 0, 3);

    // Load x directly in WMMA B layout: xv[2kk]->K0/K2 row, xv[2kk+1]->K1/K3
    float xv[16];
    #pragma unroll
    for(int kk = 0; kk < 8; kk++){
      xv[2*kk]   = xt[64*kk + base0];
      xv[2*kk+1] = xt[64*kk + base0 + 16];
    }
    // u = x + cf(m) * x[partner(m)] (partner lives in a lane of same half)
    float uv[16];
    #pragma unroll
    for(int q = 0; q < 16; q++){
      const float pz = __int_as_float(
          __builtin_amdgcn_ds_bpermute(bpidx, __float_as_int(xv[q])));
      uv[q] = fmaf(cf, pz, xv[q]);
    }

    // 4 GEMMs (jx, jy, gx, gy): D[C,m] += W[C,c] * u[c,m], K chunked by 4
    v8f acc[4][2];
    const v8f zz = {0.f,0.f,0.f,0.f,0.f,0.f,0.f,0.f};
    #pragma unroll
    for(int w = 0; w < 4; w++)
      #pragma unroll
      for(int t = 0; t < 2; t++) acc[w][t] = zz;
    #pragma unroll
    for(int kk = 0; kk < 8; kk++){
      v2f B; B[0] = uv[2*kk]; B[1] = uv[2*kk+1];
      #pragma unroll
      for(int w = 0; w < 4; w++)
        #pragma unroll
        for(int t = 0; t < 2; t++)
          acc[w][t] = __builtin_amdgcn_wmma_f32_16x16x4_f32(
              false, Wa[w][t][kk], false, B, (short)0, acc[w][t], false, false);
    }

    float* outT = out + (size_t)tok * 1024;

    // ===== geometric product path (output channels 32..63) =====
    stage_pair(slds, acc[2], acc[3], m, half);
    __asm__ volatile("" ::: "memory");
    float Xr[16], Yr[16];
    #pragma unroll
    for(int i = 0; i < 16; i++){ Xr[i] = slds[0][i][lane]; Yr[i] = slds[1][i][lane]; }
    float g[16];
    #pragma unroll
    for(int i = 0; i < 16; i++) g[i] = 0.f;
    #pragma unroll
    for(int i = 0; i < 16; i++)
      #pragma unroll
      for(int j = 0; j < 16; j++){
        const float s = pga::T.gs[i][j];
        if(s != 0.f){
          const int k = pga::T.gk[i][j];
          g[k] = fmaf(s * Xr[i], Yr[j], g[k]);
        }
      }
    {
      float4* gp = (float4*)(outT + 512 + lane*16);
      gp[0] = make_float4(g[0], g[1], g[2], g[3]);
      gp[1] = make_float4(g[4], g[5], g[6], g[7]);
      gp[2] = make_float4(g[8], g[9], g[10], g[11]);
      gp[3] = make_float4(g[12], g[13], g[14], g[15]);
    }
    __asm__ volatile("" ::: "memory");

    // ===== join path (output channels 0..31) =====
    stage_pair(slds, acc[0], acc[1], m, half);
    __asm__ volatile("" ::: "memory");
    #pragma unroll
    for(int i = 0; i < 16; i++){ Xr[i] = slds[0][i][lane]; Yr[i] = slds[1][i][lane]; }
    float dx[16], dy[16];
    #pragma unroll
    for(int j = 0; j < 16; j++){
      dx[j] = pga::T.ds[15 - j] * Xr[15 - j];
      dy[j] = pga::T.ds[15 - j] * Yr[15 - j];
    }
    float ln[16];
    #pragma unroll
    for(int i = 0; i < 16; i++) ln[i] = 0.f;
    #pragma unroll
    for(int i = 0; i < 16; i++)
      #pragma unroll
      for(int j = 0; j < 16; j++){
        const float s = pga::T.we[i][j];
        if(s != 0.f){
          const int k = pga::T.gk[i][j];
          ln[k] = fmaf(s * dx[i], dy[j], ln[k]);
        }
      }
    float dl[16];
    #pragma unroll
    for(int j = 0; j < 16; j++) dl[j] = pga::T.ds[15 - j] * ln[15 - j];
    float jn[16];
    #pragma unroll
    for(int i = 0; i < 16; i++) jn[i] = 0.f;
    #pragma unroll
    for(int j = 0; j < 16; j++){
      const float s = pga::T.gs[15][j];              // e0123 * blade_j: <=1 term
      if(s != 0.f) jn[pga::T.gk[15][j]] += s * dl[j];
    }
    {
      float4* jp = (float4*)(outT + lane*16);
      jp[0] = make_float4(z15*jn[0],  z15*jn[1],  z15*jn[2],  z15*jn[3]);
      jp[1] = make_float4(z15*jn[4],  z15*jn[5],  z15*jn[6],  z15*jn[7]);
      jp[2] = make_float4(z15*jn[8],  z15*jn[9],  z15*jn[10], z15*jn[11]);
      jp[3] = make_float4(z15*jn[12], z15*jn[13], z15*jn[14], z15*jn[15]);
    }
    __asm__ volatile("" ::: "memory");   // LDS reused next iteration
  }
}

// ---------------- launcher ---------------------------------------------------
extern "C" void kernel_launch(void* const* d_in, const int* in_sizes, int n_in,
                              void* d_out, int out_size, void* d_ws, size_t ws_size,
                              hipStream_t stream){
  const float* x   = (const float*)d_in[0];
  const float* wjx = (const float*)d_in[1];
  const float* wjy = (const float*)d_in[2];
  const float* wgx = (const float*)d_in[3];
  const float* wgy = (const float*)d_in[4];
  float* ws  = (float*)d_ws;
  float* out = (float*)d_out;

  const int nTok = in_sizes[0] / 512;          // B*S tokens (x is [tok][32][16])
  const int nRed = nTok * 32;                  // elements of x[...,15]

  pga_prep_w  <<<16, 256, 0, stream>>>(wjx, wjy, wgx, wgy, ws);
  pga_reduce1 <<<256, 256, 0, stream>>>(x, ws, nRed);
  pga_reduce2 <<<1, 256, 0, stream>>>(ws, (float)nRed);
  pga_bilinear_main<<<256, 256, 0, stream>>>(x, ws, out, nTok);
}